// Criterion_36945308680556
// MI455X (gfx1250) — compile-verified
//
#include <hip/hip_runtime.h>

typedef float v2f __attribute__((ext_vector_type(2)));
typedef float v8f __attribute__((ext_vector_type(8)));

#define BATCH   2
#define NPTS    8192
#define MOBS    8192
#define NFACES  16384
#define EPS_C   0.001f

// Workspace layout (in floats):
//   [0..15]                      : accumulator (acc at [0]), padding for alignment
//   per batch b, base = 16 + b*12*NFACES :
//     [0      .. 2F)  : plane0  (float2[F]) = (fx, fy)          -> B rows K=0,1
//     [2F     .. 4F)  : plane1  (float2[F]) = (fz, -0.5*|f|^2)  -> B rows K=2,3
//     [4F     .. 8F)  : fnext   (float4[F]) = face_next mean
//     [8F     .. 12F) : nrm     (float4[F]) = unit face normal

__global__ void zero_acc_kernel(float* ws) {
    if (threadIdx.x < 16) ws[threadIdx.x] = 0.0f;
}

__global__ void prep_faces_kernel(const float* __restrict__ ob_curr,
                                  const float* __restrict__ ob_next,
                                  const int*   __restrict__ faces,
                                  float*       __restrict__ ws) {
    int gid = blockIdx.x * blockDim.x + threadIdx.x;
    if (gid >= BATCH * NFACES) return;
    int b = gid / NFACES;
    int f = gid - b * NFACES;

    int i0 = faces[f * 3 + 0];
    int i1 = faces[f * 3 + 1];
    int i2 = faces[f * 3 + 2];

    const float* oc = ob_curr + (size_t)b * MOBS * 3;
    const float* on = ob_next + (size_t)b * MOBS * 3;
    const float inv3 = 1.0f / 3.0f;

    // face_curr mean + squared-norm bias folded into B's 4th K row
    float fcx = (oc[i0*3+0] + oc[i1*3+0] + oc[i2*3+0]) * inv3;
    float fcy = (oc[i0*3+1] + oc[i1*3+1] + oc[i2*3+1]) * inv3;
    float fcz = (oc[i0*3+2] + oc[i1*3+2] + oc[i2*3+2]) * inv3;
    float f2  = fcx*fcx + fcy*fcy + fcz*fcz;

    float* base = ws + 16 + (size_t)b * 12 * NFACES;
    float2* plane0 = (float2*)base;
    float2* plane1 = (float2*)(base + 2 * NFACES);
    float4* fnext  = (float4*)(base + 4 * NFACES);
    float4* nrm    = (float4*)(base + 8 * NFACES);

    plane0[f] = make_float2(fcx, fcy);
    plane1[f] = make_float2(fcz, -0.5f * f2);

    // face_next mean + unit normal from next obstacle positions
    float v0x = on[i0*3+0], v0y = on[i0*3+1], v0z = on[i0*3+2];
    float v1x = on[i1*3+0], v1y = on[i1*3+1], v1z = on[i1*3+2];
    float v2x = on[i2*3+0], v2y = on[i2*3+1], v2z = on[i2*3+2];

    float fnx = (v0x + v1x + v2x) * inv3;
    float fny = (v0y + v1y + v2y) * inv3;
    float fnz = (v0z + v1z + v2z) * inv3;

    float e1x = v1x - v0x, e1y = v1y - v0y, e1z = v1z - v0z;
    float e2x = v2x - v0x, e2y = v2y - v0y, e2z = v2z - v0z;
    float nx = e1y * e2z - e1z * e2y;
    float ny = e1z * e2x - e1x * e2z;
    float nz = e1x * e2y - e1y * e2x;
    float nl = sqrtf(nx*nx + ny*ny + nz*nz);
    nx /= nl; ny /= nl; nz /= nl;

    fnext[f] = make_float4(fnx, fny, fnz, 0.0f);
    nrm[f]   = make_float4(nx, ny, nz, 0.0f);
}

// One wave (32 threads) per 16-row tile of query points.
// Score s[n,f] = p_n . f_c - 0.5*|f_c|^2 ; argmin d2 == argmax s.
__global__ __launch_bounds__(32) void nn_loss_kernel(const float* __restrict__ curr,
                                                     const float* __restrict__ nxt,
                                                     float* __restrict__ ws) {
    const int tilesPerBatch = NPTS / 16;
    int tile    = blockIdx.x;
    int b       = tile / tilesPerBatch;
    int rowBase = (tile - b * tilesPerBatch) * 16;

    int lane = threadIdx.x;
    int m    = lane & 15;
    int half = lane >> 4;

    float* base = ws + 16 + (size_t)b * 12 * NFACES;
    const float2* pl = (const float2*)(base + (half ? 2 * NFACES : 0));

    // A matrix (16x4): row = rowBase+m, K-cols (x,y,z,1).
    // f32 A layout: lanes 0-15 hold K0(K1 in vgpr1), lanes 16-31 hold K2/K3.
    int row = rowBase + m;
    const float* cp = curr + ((size_t)b * NPTS + row) * 3;
    float px = cp[0], py = cp[1], pz = cp[2];
    v2f a;
    a[0] = half ? pz   : px;
    a[1] = half ? 1.0f : py;

    float best[8];
    int   bidx[8];
#pragma unroll
    for (int k = 0; k < 8; ++k) { best[k] = -3.402823466e38f; bidx[k] = 0; }

    for (int c = 0; c < NFACES; c += 16) {
        // B matrix (4x16): column N = c+m ; low half supplies K0/K1, high half K2/K3.
        float2 bv = pl[c + m];
        v2f bb;
        bb[0] = bv.x;
        bb[1] = bv.y;
        v8f cz = {};
        v8f s = __builtin_amdgcn_wmma_f32_16x16x4_f32(
            /*neg_a=*/false, a, /*neg_b=*/false, bb,
            /*c_mod=*/(short)0, cz, /*reuse_a=*/false, /*reuse_b=*/false);

        int cand = c + m;
#pragma unroll
        for (int k = 0; k < 8; ++k) {
            bool gt = s[k] > best[k];
            best[k] = gt ? s[k] : best[k];
            bidx[k] = gt ? cand : bidx[k];
        }
    }

    // Argmax reduction across the 16 lanes of each half (columns),
    // ties -> smaller face index (matches argmin's first-min semantics).
#pragma unroll
    for (int off = 1; off < 16; off <<= 1) {
#pragma unroll
        for (int k = 0; k < 8; ++k) {
            float ob = __shfl_xor(best[k], off, 32);
            int   oi = __shfl_xor(bidx[k], off, 32);
            bool take = (ob > best[k]) || (ob == best[k] && oi < bidx[k]);
            best[k] = take ? ob : best[k];
            bidx[k] = take ? oi : bidx[k];
        }
    }

    // C/D layout: slot k = row k (lanes 0-15) or row 8+k (lanes 16-31).
    float local = 0.0f;
    if (m < 8) {
        int r = rowBase + half * 8 + m;
        int f = bidx[m];
        const float4* fnext = (const float4*)(base + 4 * NFACES);
        const float4* nrm   = (const float4*)(base + 8 * NFACES);
        float4 fn = fnext[f];
        float4 nn = nrm[f];
        const float* np = nxt + ((size_t)b * NPTS + r) * 3;
        float dist = (np[0] - fn.x) * nn.x + (np[1] - fn.y) * nn.y + (np[2] - fn.z) * nn.z;
        float t = EPS_C - dist;
        local = (t > 0.0f) ? t * t * t : 0.0f;
    }

    // wave-wide sum
#pragma unroll
    for (int off = 16; off > 0; off >>= 1) local += __shfl_xor(local, off, 32);
    if (lane == 0) atomicAdd(ws, local);
}

__global__ void finalize_kernel(const float* __restrict__ ws,
                                const int* __restrict__ iter,
                                float* __restrict__ out) {
    int it = iter[0] - 50000;
    if (it < 0) it = 0;
    float prog = (float)it / 100000.0f;
    prog = fminf(prog, 1.0f);
    float w = 5000.0f + (500000.0f - 5000.0f) * prog;
    out[0] = ws[0] / (float)BATCH * w;
}

extern "C" void kernel_launch(void* const* d_in, const int* in_sizes, int n_in,
                              void* d_out, int out_size, void* d_ws, size_t ws_size,
                              hipStream_t stream) {
    const float* curr  = (const float*)d_in[0];
    const float* nxt   = (const float*)d_in[1];
    const float* oc    = (const float*)d_in[2];
    const float* on    = (const float*)d_in[3];
    const int*   faces = (const int*)d_in[4];
    const int*   iter  = (const int*)d_in[5];
    float* ws  = (float*)d_ws;
    float* out = (float*)d_out;

    zero_acc_kernel<<<1, 32, 0, stream>>>(ws);

    int totalFaces = BATCH * NFACES;
    prep_faces_kernel<<<(totalFaces + 255) / 256, 256, 0, stream>>>(oc, on, faces, ws);

    int tiles = BATCH * (NPTS / 16);
    nn_loss_kernel<<<tiles, 32, 0, stream>>>(curr, nxt, ws);

    finalize_kernel<<<1, 1, 0, stream>>>(ws, iter, out);
}